// SOMNetwork_67053029425165
// MI455X (gfx1250) — compile-verified
//
#include <hip/hip_runtime.h>
#include <hip/hip_bf16.h>
#include <math.h>

typedef float v2f __attribute__((ext_vector_type(2)));
typedef float v8f __attribute__((ext_vector_type(8)));

// D = A(16x4 f32) * B(4x16 f32) + C(16x16 f32), full-precision WMMA.
__device__ __forceinline__ v8f wmma4(v2f a, v2f b, v8f c) {
  return __builtin_amdgcn_wmma_f32_16x16x4_f32(false, a, false, b, (short)0, c,
                                               false, false);
}

#define ALPHA0 0.9f

// ------------------------------------------- K0: ||w||^2, zero-padded to Opad
__global__ void row_sqnorm_pad(const float* __restrict__ w, int O, int K,
                               int Opad, float* __restrict__ out) {
  int o = blockIdx.x * blockDim.x + threadIdx.x;
  if (o < Opad) {
    float s = 0.f;
    if (o < O) {
      const float* r = w + (size_t)o * K;
      for (int k = 0; k < K; ++k) s += r[k] * r[k];
    }
    out[o] = s;
  }
}

// ------------------------------------------------- K0b: zero-padded weights
__global__ void pad_rows(const float* __restrict__ src, int O, int K, int Opad,
                         int Kpad, float* __restrict__ dst) {
  int t = blockIdx.x * blockDim.x + threadIdx.x;
  if (t < Opad * Kpad) {
    int o = t / Kpad, k = t % Kpad;
    dst[t] = (o < O && k < K) ? src[o * K + k] : 0.f;
  }
}

// ---------------------------------------------------- K1: rbf_conv1 + sfm1
// One block per (image b, output row h). M = 24 positions, K = 25, N = 100.
// Fully padded LDS operands: branch-free WMMA loop AND branch-free epilogue.
__global__ void __launch_bounds__(256) stage1_kernel(
    const float* __restrict__ x, const float* __restrict__ w1,
    const float* __restrict__ wsq1p, const float* __restrict__ stdp,
    const float* __restrict__ thrp, float* __restrict__ out1) {
  __shared__ __align__(16) float xr[5 * 28];
  __shared__ __align__(16) float Ap[32 * 28];   // padded patches [m][k]
  __shared__ __align__(16) float Bp[112 * 28];  // padded w1 [n][k]
  __shared__ float ps[32];                      // padded row norms
  __shared__ float act[32 * 112];               // padded activations

  const int bidx = blockIdx.x;
  const int b = bidx / 24, h = bidx % 24;
  const int tid = threadIdx.x;

  if (tid < 140)
    xr[tid] = x[(size_t)b * 784 + (size_t)(h + tid / 28) * 28 + (tid % 28)];
  __syncthreads();

  // --- zero-fill pads (disjoint from valid fills; no guarded loads) ---
  for (int t = tid; t < 8 * 28; t += 256)                 // Ap rows 24..31
    Ap[24 * 28 + t] = 0.f;
  for (int t = tid; t < 24 * 3; t += 256)                 // Ap cols 25..27
    Ap[(t / 3) * 28 + 25 + (t % 3)] = 0.f;
  for (int t = tid; t < 12 * 28; t += 256)                // Bp rows 100..111
    Bp[100 * 28 + t] = 0.f;
  for (int t = tid; t < 100 * 3; t += 256)                // Bp cols 25..27
    Bp[(t / 3) * 28 + 25 + (t % 3)] = 0.f;
  // --- valid fills ---
  for (int t = tid; t < 24 * 25; t += 256) {              // patches
    const int m = t / 25, k = t % 25;
    Ap[m * 28 + k] = xr[(k / 5) * 28 + m + (k % 5)];
  }
  for (int t = tid; t < 100 * 25; t += 256) {             // w1
    Bp[(t / 25) * 28 + (t % 25)] = w1[t];
  }
  if (tid < 32) {
    float s = 0.f;
    if (tid < 24) {
#pragma unroll
      for (int ky = 0; ky < 5; ++ky)
#pragma unroll
        for (int kx = 0; kx < 5; ++kx) {
          float v = xr[ky * 28 + tid + kx];
          s += v * v;
        }
    }
    ps[tid] = s;
  }
  __syncthreads();

  const int lane = tid & 31, wave = tid >> 5;
  const int ln = lane & 15;
  const int hi2 = (lane >> 4) * 2;
  const float sd = stdp[0];
  const float inv2s2 = 1.0f / (2.0f * sd * sd);
  const float thr = thrp[0];

  for (int tile = wave; tile < 14; tile += 8) {  // 2 m-tiles x 7 n-tiles
    const int mt = tile & 1, nt = tile >> 1;
    const int n = nt * 16 + ln;
    const float wsn = wsq1p[n];  // padded: unconditional, hoisted
    const float* arow = &Ap[(mt * 16 + ln) * 28 + hi2];
    const float* brow = &Bp[n * 28 + hi2];
    v8f acc = {0.f, 0.f, 0.f, 0.f, 0.f, 0.f, 0.f, 0.f};
#pragma unroll
    for (int kk = 0; kk < 7; ++kk) {
      v2f a = *(const v2f*)(arow + kk * 4);
      v2f bb = *(const v2f*)(brow + kk * 4);
      acc = wmma4(a, bb, acc);
    }
#pragma unroll
    for (int r = 0; r < 8; ++r) {  // fully branch-free epilogue
      const int mm = mt * 16 + r + 8 * (lane >> 4);
      float sq = fmaxf(ps[mm] + wsn - 2.0f * acc[r], 0.f);
      float g = __expf(-sq * inv2s2);
      act[mm * 112 + n] = (g >= thr) ? g : 0.f;
    }
  }
  __syncthreads();

  if (tid < 100) {  // sfm1: alpha-weighted mean over the 24 columns
    float s = 0.f, coef = 1.0f;  // coef = 0.9^j for column 23-j
#pragma unroll
    for (int j = 0; j < 24; ++j) {
      s += coef * act[(23 - j) * 112 + tid];
      coef *= ALPHA0;
    }
    out1[(size_t)bidx * 100 + tid] = s * (1.0f / 24.0f);
  }
}

// ---------------------------------------------------- K2: rbf_conv2 + sfm2
// One block per image. M = 24 rows, K = 100, N = 225 (B from padded wp2).
__global__ void __launch_bounds__(256) stage2_kernel(
    const float* __restrict__ in1, const float* __restrict__ wp2,
    const float* __restrict__ wsq2p, const float* __restrict__ stdp,
    const float* __restrict__ thrp, float* __restrict__ out2) {
  __shared__ __align__(16) float Ain[32 * 100];
  __shared__ float psr[32];
  __shared__ float act[32 * 240];

  const int b = blockIdx.x;
  const int tid = threadIdx.x;
  for (int t = tid; t < 2400; t += 256) Ain[t] = in1[(size_t)b * 2400 + t];
  for (int t = tid + 2400; t < 3200; t += 256) Ain[t] = 0.f;
  __syncthreads();
  if (tid < 32) {
    float s = 0.f;
    if (tid < 24) {
      for (int k = 0; k < 100; ++k) {
        float v = Ain[tid * 100 + k];
        s += v * v;
      }
    }
    psr[tid] = s;
  }
  __syncthreads();

  const int lane = tid & 31, wave = tid >> 5;
  const int ln = lane & 15, hi2 = (lane >> 4) * 2;
  const float sd = stdp[0];
  const float inv2s2 = 1.0f / (2.0f * sd * sd);
  const float thr = thrp[0];

  for (int tile = wave; tile < 30; tile += 8) {  // 2 m-tiles x 15 n-tiles
    const int mt = tile & 1, nt = tile >> 1;
    const int n = nt * 16 + ln;
    const float wsn = wsq2p[n];
    const float* arow = &Ain[(mt * 16 + ln) * 100 + hi2];
    const float* brow = wp2 + (size_t)n * 100 + hi2;
    v8f acc = {0.f, 0.f, 0.f, 0.f, 0.f, 0.f, 0.f, 0.f};
#pragma unroll 5
    for (int kk = 0; kk < 25; ++kk) {
      v2f a = *(const v2f*)(arow + kk * 4);
      v2f bb = *(const v2f*)(brow + kk * 4);
      acc = wmma4(a, bb, acc);
    }
#pragma unroll
    for (int r = 0; r < 8; ++r) {
      const int mm = mt * 16 + r + 8 * (lane >> 4);
      float sq = fmaxf(psr[mm] + wsn - 2.0f * acc[r], 0.f);
      float g = __expf(-sq * inv2s2);
      act[mm * 240 + n] = (g >= thr) ? g : 0.f;
    }
  }
  __syncthreads();

  for (int t = tid; t < 900; t += 256) {  // sfm2: groups of 6 rows -> 4 rows
    const int g = t / 225, n = t % 225;
    float s = 0.f, coef = 1.0f;  // coef = 0.9^j for row 6g + 5-j
#pragma unroll
    for (int j = 0; j < 6; ++j) {
      s += coef * act[(6 * g + 5 - j) * 240 + n];
      coef *= ALPHA0;
    }
    out2[((size_t)b * 4 + g) * 225 + n] = s * (1.0f / 6.0f);
  }
}

// ---------------------------------------------------- K3: rbf_conv3 + sfm3
// One block per 4 images. M = 16 rows, K = 225->228, N = 625 (padded wp3).
__global__ void __launch_bounds__(256) stage3_kernel(
    const float* __restrict__ in2, const float* __restrict__ wp3,
    const float* __restrict__ wsq3p, const float* __restrict__ stdp,
    const float* __restrict__ thrp, float* __restrict__ out3) {
  __shared__ __align__(16) float Ain[16 * 228];
  __shared__ float psr[16];
  __shared__ float act[16 * 640];

  const int b0 = blockIdx.x * 4;
  const int tid = threadIdx.x;
  for (int t = tid; t < 16 * 225; t += 256) {
    const int r = t / 225, k = t % 225;
    Ain[r * 228 + k] = in2[((size_t)b0 * 4 + r) * 225 + k];
  }
  for (int t = tid; t < 16 * 3; t += 256)
    Ain[(t / 3) * 228 + 225 + (t % 3)] = 0.f;
  __syncthreads();
  if (tid < 16) {
    float s = 0.f;
    for (int k = 0; k < 225; ++k) {
      float v = Ain[tid * 228 + k];
      s += v * v;
    }
    psr[tid] = s;
  }
  __syncthreads();

  const int lane = tid & 31, wave = tid >> 5;
  const int ln = lane & 15, hi2 = (lane >> 4) * 2;
  const float sd = stdp[0];
  const float inv2s2 = 1.0f / (2.0f * sd * sd);
  const float thr = thrp[0];

  for (int tile = wave; tile < 40; tile += 8) {  // 40 n-tiles, 1 m-tile
    const int n = tile * 16 + ln;
    const float wsn = wsq3p[n];
    const float* arow = &Ain[ln * 228 + hi2];
    const float* brow = wp3 + (size_t)n * 228 + hi2;
    v8f acc = {0.f, 0.f, 0.f, 0.f, 0.f, 0.f, 0.f, 0.f};
#pragma unroll 4
    for (int kk = 0; kk < 57; ++kk) {
      v2f a = *(const v2f*)(arow + kk * 4);
      v2f bb = *(const v2f*)(brow + kk * 4);
      acc = wmma4(a, bb, acc);
    }
#pragma unroll
    for (int r = 0; r < 8; ++r) {
      const int mm = r + 8 * (lane >> 4);
      float sq = fmaxf(psr[mm] + wsn - 2.0f * acc[r], 0.f);
      float g = __expf(-sq * inv2s2);
      act[mm * 640 + n] = (g >= thr) ? g : 0.f;
    }
  }
  __syncthreads();

  for (int t = tid; t < 2500; t += 256) {  // sfm3: mean over 4 rows per image
    const int img = t / 625, c = t % 625;
    float s = 0.f, coef = 1.0f;  // coef = 0.9^j for row img*4 + 3-j
#pragma unroll
    for (int j = 0; j < 4; ++j) {
      s += coef * act[(img * 4 + 3 - j) * 640 + c];
      coef *= ALPHA0;
    }
    out3[(size_t)(b0 + img) * 625 + c] = s * 0.25f;
  }
}

// ---------------------------------------------------- K4: rbf_conv4
// One block per 16 images. M = 16, K = 625->628, N = 1225 (padded wp4).
// Output h4 is padded to stride 1232 so stores are unconditional.
__global__ void __launch_bounds__(256) stage4_kernel(
    const float* __restrict__ in3, const float* __restrict__ wp4,
    const float* __restrict__ wsq4p, const float* __restrict__ stdp,
    const float* __restrict__ thrp, float* __restrict__ h4) {
  __shared__ __align__(16) float Ain[16 * 628];
  __shared__ float psr[16];

  const int b0 = blockIdx.x * 16;
  const int tid = threadIdx.x;
  for (int t = tid; t < 16 * 625; t += 256) {
    const int r = t / 625, k = t % 625;
    Ain[r * 628 + k] = in3[(size_t)(b0 + r) * 625 + k];
  }
  for (int t = tid; t < 16 * 3; t += 256)
    Ain[(t / 3) * 628 + 625 + (t % 3)] = 0.f;
  __syncthreads();
  if (tid < 16) {
    float s = 0.f;
    for (int k = 0; k < 625; ++k) {
      float v = Ain[tid * 628 + k];
      s += v * v;
    }
    psr[tid] = s;
  }
  __syncthreads();

  const int lane = tid & 31, wave = tid >> 5;
  const int ln = lane & 15, hi2 = (lane >> 4) * 2;
  const float sd = stdp[0];
  const float inv2s2 = 1.0f / (2.0f * sd * sd);
  const float thr = thrp[0];

  for (int tile = wave; tile < 77; tile += 8) {  // 77 n-tiles
    const int n = tile * 16 + ln;
    const float wsn = wsq4p[n];
    const float* arow = &Ain[ln * 628 + hi2];
    const float* brow = wp4 + (size_t)n * 628 + hi2;
    v8f acc = {0.f, 0.f, 0.f, 0.f, 0.f, 0.f, 0.f, 0.f};
#pragma unroll 4
    for (int kk = 0; kk < 157; ++kk) {
      v2f a = *(const v2f*)(arow + kk * 4);
      v2f bb = *(const v2f*)(brow + kk * 4);
      acc = wmma4(a, bb, acc);
    }
#pragma unroll
    for (int r = 0; r < 8; ++r) {
      const int mm = r + 8 * (lane >> 4);
      float sq = fmaxf(psr[mm] + wsn - 2.0f * acc[r], 0.f);
      float g = __expf(-sq * inv2s2);
      h4[(size_t)(b0 + mm) * 1232 + n] = (g >= thr) ? g : 0.f;
    }
  }
}

// ---------------------------------------------------- K5: FC + softmax
__global__ void __launch_bounds__(128) fc_softmax_kernel(
    const float* __restrict__ h4, const float* __restrict__ fcw,
    const float* __restrict__ fcb, float* __restrict__ out) {
  __shared__ float red[10][128];
  const int b = blockIdx.x, t = threadIdx.x;
  float loc[10];
#pragma unroll
  for (int j = 0; j < 10; ++j) loc[j] = 0.f;
  for (int k = t; k < 1225; k += 128) {
    float hv = h4[(size_t)b * 1232 + k];
#pragma unroll
    for (int j = 0; j < 10; ++j) loc[j] += hv * fcw[k * 10 + j];
  }
#pragma unroll
  for (int j = 0; j < 10; ++j) red[j][t] = loc[j];
  __syncthreads();
  for (int s = 64; s > 0; s >>= 1) {
    if (t < s) {
#pragma unroll
      for (int j = 0; j < 10; ++j) red[j][t] += red[j][t + s];
    }
    __syncthreads();
  }
  if (t == 0) {
    float lg[10], mx = -1e30f;
#pragma unroll
    for (int j = 0; j < 10; ++j) {
      lg[j] = red[j][0] + fcb[j];
      mx = fmaxf(mx, lg[j]);
    }
    float sm = 0.f;
#pragma unroll
    for (int j = 0; j < 10; ++j) {
      lg[j] = __expf(lg[j] - mx);
      sm += lg[j];
    }
    float inv = 1.f / sm;
#pragma unroll
    for (int j = 0; j < 10; ++j) out[b * 10 + j] = lg[j] * inv;
  }
}

extern "C" void kernel_launch(void* const* d_in, const int* in_sizes, int n_in,
                              void* d_out, int out_size, void* d_ws,
                              size_t ws_size, hipStream_t stream) {
  const float* x = (const float*)d_in[0];
  const float* w1 = (const float*)d_in[1];
  const float* w2 = (const float*)d_in[2];
  const float* w3 = (const float*)d_in[3];
  const float* w4 = (const float*)d_in[4];
  const float* std1 = (const float*)d_in[5];
  const float* std2 = (const float*)d_in[6];
  const float* std3 = (const float*)d_in[7];
  const float* std4 = (const float*)d_in[8];
  const float* b1 = (const float*)d_in[9];
  const float* b2 = (const float*)d_in[10];
  const float* b3 = (const float*)d_in[11];
  const float* b4 = (const float*)d_in[12];
  const float* fcw = (const float*)d_in[13];
  const float* fcb = (const float*)d_in[14];

  float* ws = (float*)d_ws;
  float* wsq1p = ws + 0;                       // 112
  float* wsq2p = ws + 128;                     // 240
  float* wsq3p = ws + 384;                     // 640
  float* wsq4p = ws + 1024;                    // 1232
  float* wp2 = ws + 4096;                      // 240 x 100 = 24000
  float* wp3 = ws + 28160;                     // 640 x 228 = 145920
  float* wp4 = ws + 174080;                    // 1232 x 628 = 773696
  float* out1 = ws + 947776;                   // 24576 x 100
  float* out2 = out1 + (size_t)24576 * 100;    // 4096 x 225
  float* out3 = out2 + (size_t)4096 * 225;     // 1024 x 625
  float* h4 = out3 + (size_t)1024 * 625;       // 1024 x 1232 (padded)

  row_sqnorm_pad<<<1, 128, 0, stream>>>(w1, 100, 25, 112, wsq1p);
  row_sqnorm_pad<<<1, 256, 0, stream>>>(w2, 225, 100, 240, wsq2p);
  row_sqnorm_pad<<<3, 256, 0, stream>>>(w3, 625, 225, 640, wsq3p);
  row_sqnorm_pad<<<5, 256, 0, stream>>>(w4, 1225, 625, 1232, wsq4p);
  pad_rows<<<(240 * 100 + 255) / 256, 256, 0, stream>>>(w2, 225, 100, 240, 100,
                                                        wp2);
  pad_rows<<<(640 * 228 + 255) / 256, 256, 0, stream>>>(w3, 625, 225, 640, 228,
                                                        wp3);
  pad_rows<<<(1232 * 628 + 255) / 256, 256, 0, stream>>>(w4, 1225, 625, 1232,
                                                         628, wp4);

  stage1_kernel<<<24576, 256, 0, stream>>>(x, w1, wsq1p, std1, b1, out1);
  stage2_kernel<<<1024, 256, 0, stream>>>(out1, wp2, wsq2p, std2, b2, out2);
  stage3_kernel<<<256, 256, 0, stream>>>(out2, wp3, wsq3p, std3, b3, out3);
  stage4_kernel<<<64, 256, 0, stream>>>(out3, wp4, wsq4p, std4, b4, h4);
  fc_softmax_kernel<<<1024, 128, 0, stream>>>(h4, fcw, fcb, (float*)d_out);
}